// Model_48318382080579
// MI455X (gfx1250) — compile-verified
//
#include <hip/hip_runtime.h>
#include <math.h>

// ---------------------------------------------------------------------------
// Problem constants (match reference)
// ---------------------------------------------------------------------------
#define BB 1024
#define DD 512
#define VV 128
#define TT 201
#define GG 2048   // 4*D

typedef __attribute__((ext_vector_type(2))) float v2f;
typedef __attribute__((ext_vector_type(8))) float v8f;

static __device__ __forceinline__ v8f wmma_f32_k4(v2f a, v2f b, v8f c) {
    // D = A(16x4,f32) * B(4x16,f32) + C(16x16,f32)  -> v_wmma_f32_16x16x4_f32
    return __builtin_amdgcn_wmma_f32_16x16x4_f32(
        false, a, false, b, (short)0, c, false, false);
}

static __device__ __forceinline__ float sigmf(float x) {
    return 1.0f / (1.0f + __expf(-x));
}

// ---------------------------------------------------------------------------
// One-time: h0 = LayerNorm(pooled)*gamma+beta ; c0 = 0 ; tok0 = sos
// ---------------------------------------------------------------------------
__global__ __launch_bounds__(256) void init_state_kernel(
    const float* __restrict__ pooled, const float* __restrict__ gamma,
    const float* __restrict__ beta, const int* __restrict__ sos_idx,
    float* __restrict__ h, float* __restrict__ c, int* __restrict__ tok) {
    __shared__ float red[256];
    const int b = blockIdx.x, tid = threadIdx.x;
    const float v0 = pooled[b * DD + tid];
    const float v1 = pooled[b * DD + 256 + tid];
    red[tid] = v0 + v1;
    __syncthreads();
    for (int s = 128; s > 0; s >>= 1) {
        if (tid < s) red[tid] += red[tid + s];
        __syncthreads();
    }
    const float mean = red[0] * (1.0f / DD);
    __syncthreads();
    const float d0 = v0 - mean, d1 = v1 - mean;
    red[tid] = d0 * d0 + d1 * d1;
    __syncthreads();
    for (int s = 128; s > 0; s >>= 1) {
        if (tid < s) red[tid] += red[tid + s];
        __syncthreads();
    }
    const float rs = rsqrtf(red[0] * (1.0f / DD) + 1e-5f);
    h[b * DD + tid]       = d0 * rs * gamma[tid]       + beta[tid];
    h[b * DD + 256 + tid] = d1 * rs * gamma[256 + tid] + beta[256 + tid];
    c[b * DD + tid] = 0.0f;
    c[b * DD + 256 + tid] = 0.0f;
    if (tid == 0) tok[b] = sos_idx[0];
}

// ---------------------------------------------------------------------------
// One-time: time-invariant attention context.
// softmax(fm@wf + const_k) == softmax(fm@wf)  ->  ctx[b,:] = sum_k aw[k]*fm[b,k,:]
// fm[b,k,d] = feature_map[b,d,k] (k = 7*7 flat)
// ---------------------------------------------------------------------------
__global__ __launch_bounds__(256) void attn_ctx_kernel(
    const float* __restrict__ fmap, const float* __restrict__ attn_w,
    float* __restrict__ ctx) {
    __shared__ float wf[DD];
    __shared__ float sc[49];
    __shared__ float aw[49];
    const int b = blockIdx.x, tid = threadIdx.x;
    wf[tid] = attn_w[tid];
    wf[tid + 256] = attn_w[tid + 256];
    __syncthreads();
    const float* fb = fmap + (size_t)b * DD * 49;
    if (tid < 49) {
        float s = 0.0f;
        for (int d = 0; d < DD; ++d) s += fb[d * 49 + tid] * wf[d];
        sc[tid] = s;
    }
    __syncthreads();
    if (tid < 49) {
        float m = sc[0];
        for (int k = 1; k < 49; ++k) m = fmaxf(m, sc[k]);
        float den = 0.0f;
        for (int k = 0; k < 49; ++k) den += __expf(sc[k] - m);
        aw[tid] = __expf(sc[tid] - m) / den;
    }
    __syncthreads();
    for (int d = tid; d < DD; d += 256) {
        const float* row = fb + (size_t)d * 49;
        float s = 0.0f;
        for (int k = 0; k < 49; ++k) s += aw[k] * row[k];
        ctx[b * DD + d] = s;
    }
}

// ---------------------------------------------------------------------------
// One-time weight prep:
//  Wt2[k][(d<<2)|g] = W_hh[g*512+d][k]          (512 x 2048, gate-interleaved)
//  XWb[v][(d<<2)|g] = embed[v]@W_ih[g*512+d] + b_ih[..] + b_hh[..]  (128 x 2048)
//  Pt [k][v]        = proj_w[v][k]              (512 x 128)
// ---------------------------------------------------------------------------
__global__ __launch_bounds__(256) void build_wt2_kernel(
    const float* __restrict__ W_hh, float* __restrict__ Wt2) {
    const int idx = blockIdx.x * 256 + threadIdx.x;  // < 512*2048
    const int k = idx >> 11, n = idx & (GG - 1);
    const int d = n >> 2, g = n & 3;
    Wt2[idx] = W_hh[(size_t)(g * DD + d) * DD + k];
}

__global__ __launch_bounds__(256) void build_xwb_kernel(
    const float* __restrict__ embed, const float* __restrict__ W_ih,
    const float* __restrict__ b_ih, const float* __restrict__ b_hh,
    float* __restrict__ XWb) {
    const int idx = blockIdx.x * 256 + threadIdx.x;  // < 128*2048
    const int v = idx >> 11, n = idx & (GG - 1);
    const int d = n >> 2, g = n & 3;
    const int row = g * DD + d;
    const float* e = embed + (size_t)v * DD;
    const float* w = W_ih + (size_t)row * DD;
    float s = b_ih[row] + b_hh[row];
    for (int k = 0; k < DD; ++k) s += e[k] * w[k];
    XWb[idx] = s;
}

__global__ __launch_bounds__(256) void build_pt_kernel(
    const float* __restrict__ proj_w, float* __restrict__ Pt) {
    const int idx = blockIdx.x * 256 + threadIdx.x;  // < 512*128
    const int k = idx >> 7, v = idx & (VV - 1);
    Pt[idx] = proj_w[v * DD + k];
}

// ---------------------------------------------------------------------------
// Per-step fused kernel:  acc = (h+ctx) @ Wt2  (M=1024, N=2048 interleaved, K=512)
// then, since each 64-wide N-tile holds all 4 gates for 16 d-columns, finish the
// LSTM update in the epilogue: gates = acc + XWb[tok[b]] ; update c, h_out.
// block = 256 threads (8 waves), tile 128(M) x 64(N), K-chunks of 32.
// wave w: N-subtile (w&3), M-subtiles (w>>2)*4 .. +3  -> 4 accumulators.
// ---------------------------------------------------------------------------
#define K1_TM 128
#define K1_TN 64
#define K1_KC 32
#define K1_LD (K1_KC + 2)
#define K1_GLD 68  // epilogue gate-tile row stride (16B multiple)

__global__ __launch_bounds__(256) void gates_lstm_kernel(
    const float* __restrict__ hin, const float* __restrict__ ctx,
    const float* __restrict__ Wt2, const float* __restrict__ XWb,
    const int* __restrict__ tok, float* __restrict__ c,
    float* __restrict__ hout) {
    // LDS reused: GEMM tiles during k-loop, gate tile in epilogue.
    __shared__ float smem[K1_TM * K1_GLD];  // 128*68 floats = 34.8 KB
    float (*As)[K1_LD] = (float(*)[K1_LD])smem;                       // 128 x 34
    float (*Bs)[K1_LD] = (float(*)[K1_LD])(smem + K1_TM * K1_LD);     // 64 x 34

    const int tid = threadIdx.x;
    const int lane = tid & 31;
    const int wave = tid >> 5;
    const int m0 = blockIdx.y * K1_TM;
    const int n0 = blockIdx.x * K1_TN;   // interleaved col base = d0*4
    const int d0 = blockIdx.x * 16;      // 16 d-columns per block
    const int wn = (wave & 3) * 16;
    const int wm = (wave >> 2) * 64;
    const int lm = lane & 15;
    const int kp = (lane >> 4) * 2;      // ISA 16x4 f32 fragment K-pair offset

    v8f acc[4];
    #pragma unroll
    for (int r = 0; r < 4; ++r) acc[r] = (v8f){0,0,0,0,0,0,0,0};

    for (int k0 = 0; k0 < DD; k0 += K1_KC) {
        // ---- stage A: 128 x 32 of (h + ctx) ----
        {
            const int row = tid >> 3;
            const int kk = (tid & 7) * 4;
            #pragma unroll
            for (int rr = 0; rr < K1_TM; rr += 32) {
                const int m = m0 + row + rr;
                const float4 hv = *(const float4*)(hin + (size_t)m * DD + k0 + kk);
                const float4 cv = *(const float4*)(ctx + (size_t)m * DD + k0 + kk);
                As[row + rr][kk + 0] = hv.x + cv.x;
                As[row + rr][kk + 1] = hv.y + cv.y;
                As[row + rr][kk + 2] = hv.z + cv.z;
                As[row + rr][kk + 3] = hv.w + cv.w;
            }
        }
        // ---- stage B: 32 x 64 from Wt2, transposed into Bs[n][k] ----
        {
            const int k = tid >> 4;
            const int n = (tid & 15) * 4;
            #pragma unroll
            for (int kk2 = 0; kk2 < K1_KC; kk2 += 16) {
                const float4 v =
                    *(const float4*)(Wt2 + (size_t)(k0 + k + kk2) * GG + n0 + n);
                Bs[n + 0][k + kk2] = v.x;
                Bs[n + 1][k + kk2] = v.y;
                Bs[n + 2][k + kk2] = v.z;
                Bs[n + 3][k + kk2] = v.w;
            }
        }
        __syncthreads();
        #pragma unroll
        for (int kk = 0; kk < K1_KC; kk += 4) {
            const v2f bf = *(const v2f*)(&Bs[wn + lm][kk + kp]);
            #pragma unroll
            for (int r = 0; r < 4; ++r) {
                const v2f af = *(const v2f*)(&As[wm + r * 16 + lm][kk + kp]);
                acc[r] = wmma_f32_k4(af, bf, acc[r]);
            }
        }
        __syncthreads();
    }

    // ---- epilogue: stage accumulators to LDS gate tile (alias As/Bs) ----
    float (*Gt)[K1_GLD] = (float(*)[K1_GLD])smem;  // 128 x 68
    #pragma unroll
    for (int r = 0; r < 4; ++r) {
        const int mbase = wm + r * 16 + (lane >> 4) * 8;
        const int col = wn + lm;
        #pragma unroll
        for (int e = 0; e < 8; ++e) {
            Gt[mbase + e][col] = acc[r][e];
        }
    }
    __syncthreads();

    // ---- fused LSTM pointwise update: 128 rows x 16 d-cols = 2048 pairs ----
    for (int p = tid; p < K1_TM * 16; p += 256) {
        const int m = p >> 4, dl = p & 15;
        const int gm = m0 + m;
        const int gd = d0 + dl;
        const int t = tok[gm];
        const float4 gv = *(const float4*)(&Gt[m][dl * 4]);          // i,f,g,o
        const float4 xw = *(const float4*)(XWb + (size_t)t * GG + (size_t)gd * 4);
        const float gi = gv.x + xw.x;
        const float gf = gv.y + xw.y;
        const float gg = gv.z + xw.z;
        const float go = gv.w + xw.w;
        const size_t ci = (size_t)gm * DD + gd;
        const float cn = sigmf(gf) * c[ci] + sigmf(gi) * tanhf(gg);
        c[ci] = cn;
        hout[ci] = sigmf(go) * tanhf(cn);
    }
}

// ---------------------------------------------------------------------------
// Per-step: logits = h @ Pt + proj_b ; out[:, :, t] = logits ; tok = argmax
// block = 256 threads (8 waves), 32 rows x full V=128, K-chunks of 32.
// wave w: M-subtile (w&1), N-subtiles (w>>1)*2 .. +1  -> 2 accumulators.
// ---------------------------------------------------------------------------
#define K3_TM 32
#define K3_KC 32
#define K3_LD (K3_KC + 2)

__global__ __launch_bounds__(256) void logits_kernel(
    const float* __restrict__ h, const float* __restrict__ Pt,
    const float* __restrict__ proj_b, float* __restrict__ out,
    int* __restrict__ tok, int tstep) {
    __shared__ float As[K3_TM][K3_LD];
    __shared__ float Bs[VV][K3_LD];
    __shared__ float Lt[K3_TM][VV + 1];

    const int tid = threadIdx.x;
    const int lane = tid & 31;
    const int wave = tid >> 5;
    const int m0 = blockIdx.x * K3_TM;
    const int ms = (wave & 1) * 16;
    const int ng = (wave >> 1) * 32;
    const int lm = lane & 15;
    const int kp = (lane >> 4) * 2;

    v8f acc[2];
    acc[0] = (v8f){0,0,0,0,0,0,0,0};
    acc[1] = (v8f){0,0,0,0,0,0,0,0};

    for (int k0 = 0; k0 < DD; k0 += K3_KC) {
        {   // stage A: 32 x 32 from h (exactly 1 float4 per thread)
            const int row = tid >> 3;
            const int kk = (tid & 7) * 4;
            const float4 v = *(const float4*)(h + (size_t)(m0 + row) * DD + k0 + kk);
            As[row][kk + 0] = v.x;
            As[row][kk + 1] = v.y;
            As[row][kk + 2] = v.z;
            As[row][kk + 3] = v.w;
        }
        {   // stage B: 32 x 128 from Pt, transposed into Bs[n][k]
            const int k = tid >> 5;          // 0..7
            const int n = (tid & 31) * 4;    // 0..124
            #pragma unroll
            for (int kk2 = 0; kk2 < K3_KC; kk2 += 8) {
                const float4 v = *(const float4*)(Pt + (size_t)(k0 + k + kk2) * VV + n);
                Bs[n + 0][k + kk2] = v.x;
                Bs[n + 1][k + kk2] = v.y;
                Bs[n + 2][k + kk2] = v.z;
                Bs[n + 3][k + kk2] = v.w;
            }
        }
        __syncthreads();
        #pragma unroll
        for (int kk = 0; kk < K3_KC; kk += 4) {
            const v2f af = *(const v2f*)(&As[ms + lm][kk + kp]);
            #pragma unroll
            for (int j = 0; j < 2; ++j) {
                const v2f bf = *(const v2f*)(&Bs[ng + j * 16 + lm][kk + kp]);
                acc[j] = wmma_f32_k4(af, bf, acc[j]);
            }
        }
        __syncthreads();
    }
    // logits tile + bias -> LDS
    #pragma unroll
    for (int j = 0; j < 2; ++j) {
        const int col = ng + j * 16 + lm;
        const float pb = proj_b[col];
        const int rbase = ms + (lane >> 4) * 8;
        #pragma unroll
        for (int e = 0; e < 8; ++e) {
            Lt[rbase + e][col] = acc[j][e] + pb;
        }
    }
    __syncthreads();
    // write logits to out (B, V, T)
    #pragma unroll
    for (int i = 0; i < (K3_TM * VV) / 256; ++i) {
        const int idx = tid + i * 256;
        const int row = idx >> 7, col = idx & (VV - 1);
        out[(size_t)((m0 + row) * VV + col) * TT + tstep] = Lt[row][col];
    }
    // first-max argmax per row -> next token
    if (tid < K3_TM) {
        float best = Lt[tid][0];
        int bi = 0;
        for (int v = 1; v < VV; ++v) {
            const float vv = Lt[tid][v];
            if (vv > best) { best = vv; bi = v; }
        }
        tok[m0 + tid] = bi;
    }
}

// ---------------------------------------------------------------------------
// Host entry
// ---------------------------------------------------------------------------
extern "C" void kernel_launch(void* const* d_in, const int* in_sizes, int n_in,
                              void* d_out, int out_size, void* d_ws, size_t ws_size,
                              hipStream_t stream) {
    const float* fmap   = (const float*)d_in[0];
    const float* pooled = (const float*)d_in[1];
    const float* gamma  = (const float*)d_in[2];
    const float* beta   = (const float*)d_in[3];
    const float* W_ih   = (const float*)d_in[4];
    const float* W_hh   = (const float*)d_in[5];
    const float* b_ih   = (const float*)d_in[6];
    const float* b_hh   = (const float*)d_in[7];
    const float* embed  = (const float*)d_in[8];
    const float* attn_w = (const float*)d_in[9];
    // d_in[10] = attn_b: constant over the spatial axis -> cancels in softmax
    const float* proj_w = (const float*)d_in[11];
    const float* proj_b = (const float*)d_in[12];
    const int*   sos    = (const int*)d_in[13];
    float* out = (float*)d_out;

    float* ws = (float*)d_ws;
    float* ha  = ws; ws += BB * DD;       // h ping
    float* hb  = ws; ws += BB * DD;       // h pong
    float* c   = ws; ws += BB * DD;
    float* ctx = ws; ws += BB * DD;
    float* Wt2 = ws; ws += DD * GG;       // gate-interleaved W_hh^T
    float* XWb = ws; ws += VV * GG;       // embed @ W_ih^T + biases (interleaved)
    float* Pt  = ws; ws += DD * VV;       // proj_w^T
    int*   tok = (int*)ws;                // current token per batch row

    // One-time setup
    init_state_kernel<<<BB, 256, 0, stream>>>(pooled, gamma, beta, sos, ha, c, tok);
    attn_ctx_kernel<<<BB, 256, 0, stream>>>(fmap, attn_w, ctx);
    build_wt2_kernel<<<(DD * GG) / 256, 256, 0, stream>>>(W_hh, Wt2);
    build_xwb_kernel<<<(VV * GG) / 256, 256, 0, stream>>>(embed, W_ih, b_ih, b_hh, XWb);
    build_pt_kernel<<<(DD * VV) / 256, 256, 0, stream>>>(proj_w, Pt);

    // Sequential recurrence: 2 kernels per step, h ping-ponged
    float* hin = ha;
    float* hnx = hb;
    for (int t = 0; t < TT; ++t) {
        gates_lstm_kernel<<<dim3(GG / K1_TN, BB / K1_TM), 256, 0, stream>>>(
            hin, ctx, Wt2, XWb, tok, c, hnx);
        logits_kernel<<<BB / K3_TM, 256, 0, stream>>>(hnx, Pt, proj_b, out, tok, t);
        float* tmp = hin; hin = hnx; hnx = tmp;
    }
}